// StochSelfAttention_51745765982557
// MI455X (gfx1250) — compile-verified
//
#include <hip/hip_runtime.h>
#include <hip/hip_bf16.h>

// Problem constants from the reference
#define Bn   32
#define Sn   576
#define Dn   768
#define Gn   24
#define MROWS (Bn * Sn)        // 18432
#define TEMP 0.01f

typedef __attribute__((ext_vector_type(2))) float v2f;
typedef __attribute__((ext_vector_type(8))) float v8f;

// ---------------------------------------------------------------------------
// Kernel 1: Q/K/V = x @ W + b via V_WMMA_F32_16X16X4_F32 (exact fp32 matmul).
// 256 threads = 8 wave32. Block tile 32M x 128N, K-chunk 64.
// x tile (32x64, LDS stride 68: bank-conflict-free + 16B aligned) and
// W tile (64x128) staged with float4 (b128) cooperative loads.
// Waves arranged 2(M) x 4(N); each wave holds TWO 16x16 accumulators and
// reuses one A fragment per K-step -> 32 WMMAs per wave per K-chunk.
// ---------------------------------------------------------------------------
__global__ __launch_bounds__(256) void qkv_wmma_kernel(
    const float* __restrict__ x,
    const float* __restrict__ Wq, const float* __restrict__ bq,
    const float* __restrict__ Wk, const float* __restrict__ bk,
    const float* __restrict__ Wv, const float* __restrict__ bv,
    float* __restrict__ Q, float* __restrict__ K, float* __restrict__ V)
{
    const float* W;  const float* bias;  float* out;
    if (blockIdx.z == 0)      { W = Wq; bias = bq; out = Q; }
    else if (blockIdx.z == 1) { W = Wk; bias = bk; out = K; }
    else                      { W = Wv; bias = bv; out = V; }

    const int m0   = blockIdx.x * 32;
    const int n0   = blockIdx.y * 128;
    const int wave = threadIdx.x >> 5;
    const int lane = threadIdx.x & 31;
    const int wm   = wave >> 2;              // 0..1 : M sub-tile
    const int wn   = wave & 3;               // 0..3 : N sub-tile (32 cols)

    __shared__ float lds_x[32 * 68];         // stride 68: 4*r banks, f4-aligned
    __shared__ float lds_w[64 * 128];

    v8f acc0 = {}, acc1 = {};

    const int mrow  = lane & 15;
    const int khalf = (lane >> 4) * 2;       // lanes 16-31 own K=2,3 of the quad
    const int arow  = wm * 16 + mrow;        // A row within x tile
    const int bc0   = wn * 32 + mrow;        // B col within W tile (acc0)
    const int bc1   = bc0 + 16;              // (acc1)

    for (int kc = 0; kc < Dn; kc += 64) {
        // x tile: 32 rows x 64 K = 512 float4 (2 per thread)
        #pragma unroll
        for (int t = 0; t < 2; ++t) {
            const int idx = threadIdx.x + t * 256;   // 0..511
            const int r   = idx >> 4;                // 16 float4 per row
            const int c4  = idx & 15;
            const float4 val =
                *(const float4*)(x + (size_t)(m0 + r) * Dn + kc + c4 * 4);
            *(float4*)(&lds_x[r * 68 + c4 * 4]) = val;
        }
        // W tile: 64 K x 128 N = 2048 float4 (8 per thread)
        #pragma unroll
        for (int t = 0; t < 8; ++t) {
            const int idx = threadIdx.x + t * 256;   // 0..2047
            const int r   = idx >> 5;                // 32 float4 per row
            const int c4  = idx & 31;
            const float4 val =
                *(const float4*)(W + (size_t)(kc + r) * Dn + n0 + c4 * 4);
            *(float4*)(&lds_w[r * 128 + c4 * 4]) = val;
        }
        __syncthreads();

        #pragma unroll
        for (int kk = 0; kk < 64; kk += 4) {
            v2f a, b0, b1;
            a.x  = lds_x[arow * 68 + kk + khalf];
            a.y  = lds_x[arow * 68 + kk + khalf + 1];
            b0.x = lds_w[(kk + khalf)     * 128 + bc0];
            b0.y = lds_w[(kk + khalf + 1) * 128 + bc0];
            b1.x = lds_w[(kk + khalf)     * 128 + bc1];
            b1.y = lds_w[(kk + khalf + 1) * 128 + bc1];
            acc0 = __builtin_amdgcn_wmma_f32_16x16x4_f32(
                       false, a, false, b0, (short)0, acc0, false, false);
            acc1 = __builtin_amdgcn_wmma_f32_16x16x4_f32(
                       false, a, false, b1, (short)0, acc1, false, false);
        }
        __syncthreads();
    }

    // C/D layout: VGPR r -> row (r + 8*(lane>=16)), col = lane&15
    const int col0  = n0 + wn * 32 + (lane & 15);
    const int rbase = m0 + wm * 16 + ((lane >> 4) << 3);
    const float bci0 = bias[col0];
    const float bci1 = bias[col0 + 16];
    #pragma unroll
    for (int r = 0; r < 8; ++r) {
        out[(size_t)(rbase + r) * Dn + col0]      = acc0[r] + bci0;
        out[(size_t)(rbase + r) * Dn + col0 + 16] = acc1[r] + bci1;
    }
}

// ---------------------------------------------------------------------------
// Kernel 2a: fc1 of key_finder_net as a WMMA GEMM:
//   H1 = relu( concat(Q,K) @ W1 + b1 )      (18432 x 1536)(1536 x 60)
// N padded to 64 in LDS (pad cols zero-filled once). Block tile 32M x 64N.
// Since K-chunks (64) never straddle the 768 boundary, each chunk reads
// entirely from Q or entirely from K.
// ---------------------------------------------------------------------------
__global__ __launch_bounds__(256) void fc1_wmma_kernel(
    const float* __restrict__ Q, const float* __restrict__ K,
    const float* __restrict__ W1, const float* __restrict__ b1,
    float* __restrict__ H1)                  // (MROWS, 60)
{
    const int m0   = blockIdx.x * 32;
    const int wave = threadIdx.x >> 5;
    const int lane = threadIdx.x & 31;
    const int wm   = wave >> 2;              // 0..1
    const int wn   = wave & 3;               // 0..3 (16-col slices of padded 64)

    __shared__ float lds_x[32 * 68];
    __shared__ float lds_w[64 * 64];

    // zero the pad columns 60..63 once (inner loop never writes them)
    lds_w[(threadIdx.x >> 2) * 64 + 60 + (threadIdx.x & 3)] = 0.0f;

    v8f acc = {};
    const int mrow  = lane & 15;
    const int khalf = (lane >> 4) * 2;
    const int arow  = wm * 16 + mrow;
    const int bcol  = wn * 16 + mrow;

    for (int kc = 0; kc < 2 * Dn; kc += 64) {
        const float* src = (kc < Dn) ? (Q + (size_t)m0 * Dn + kc)
                                     : (K + (size_t)m0 * Dn + (kc - Dn));
        // A tile: 32 x 64 = 512 float4
        #pragma unroll
        for (int t = 0; t < 2; ++t) {
            const int idx = threadIdx.x + t * 256;
            const int r   = idx >> 4;
            const int c4  = idx & 15;
            const float4 val = *(const float4*)(src + (size_t)r * Dn + c4 * 4);
            *(float4*)(&lds_x[r * 68 + c4 * 4]) = val;
        }
        // W1 tile: 64 rows x 60 cols = 960 float4 (row stride 60, 60 % 4 == 0)
        for (int idx = threadIdx.x; idx < 960; idx += 256) {
            const int r  = idx / 15;                 // 15 float4 per row
            const int c4 = idx - r * 15;
            const float4 val =
                *(const float4*)(W1 + (size_t)(kc + r) * 60 + c4 * 4);
            *(float4*)(&lds_w[r * 64 + c4 * 4]) = val;
        }
        __syncthreads();

        #pragma unroll
        for (int kk = 0; kk < 64; kk += 4) {
            v2f a, b;
            a.x = lds_x[arow * 68 + kk + khalf];
            a.y = lds_x[arow * 68 + kk + khalf + 1];
            b.x = lds_w[(kk + khalf)     * 64 + bcol];
            b.y = lds_w[(kk + khalf + 1) * 64 + bcol];
            acc = __builtin_amdgcn_wmma_f32_16x16x4_f32(
                      false, a, false, b, (short)0, acc, false, false);
        }
        __syncthreads();
    }

    const int col   = wn * 16 + (lane & 15);         // 0..63
    const int rbase = m0 + wm * 16 + ((lane >> 4) << 3);
    if (col < 60) {
        const float bc = b1[col];
        #pragma unroll
        for (int r = 0; r < 8; ++r)
            H1[(size_t)(rbase + r) * 60 + col] = fmaxf(acc[r] + bc, 0.0f);
    }
}

// ---------------------------------------------------------------------------
// Kernel 2b: fc2 -> relu -> fc3 -> tanh (tiny). One 64-thread block per row.
// ---------------------------------------------------------------------------
__global__ __launch_bounds__(64) void mlp_tail_kernel(
    const float* __restrict__ H1,
    const float* __restrict__ W2, const float* __restrict__ b2,
    const float* __restrict__ W3, const float* __restrict__ b3,
    float* __restrict__ sample)              // (MROWS, 2)
{
    const int row = blockIdx.x;
    __shared__ float h1[60];
    __shared__ float h2[60];

    if (threadIdx.x < 60)
        h1[threadIdx.x] = H1[(size_t)row * 60 + threadIdx.x];
    __syncthreads();

    const int j = threadIdx.x;
    if (j < 60) {
        float acc = b2[j];
        #pragma unroll
        for (int i = 0; i < 60; ++i)
            acc = fmaf(h1[i], W2[i * 60 + j], acc);
        h2[j] = fmaxf(acc, 0.0f);
    }
    __syncthreads();
    if (j < 2) {
        float acc = b3[j];
        #pragma unroll
        for (int i = 0; i < 60; ++i)
            acc = fmaf(h2[i], W3[i * 2 + j], acc);
        sample[(size_t)row * 2 + j] = tanhf(acc);
    }
}

// ---------------------------------------------------------------------------
// Kernel 3: bilinear gather of K/V at the sampled grid point,
// score = q . sampled_k, out = sigmoid(TEMP*score) * sampled_v.
// 192 threads per (b,s); one float4 (4 channels) per thread -> b128 loads.
// ---------------------------------------------------------------------------
__global__ __launch_bounds__(192) void sample_attend_kernel(
    const float* __restrict__ Q, const float* __restrict__ K,
    const float* __restrict__ V, const float* __restrict__ sample,
    float* __restrict__ out)
{
    const int s = blockIdx.x;          // 0..575
    const int b = blockIdx.y;          // 0..31

    // grid = concat(sample[:,:,0], sample[:,:,1], dim=1).reshape(B,G,G,2)
    const int t0 = 2 * s, t1 = 2 * s + 1;
    const float gx = (t0 < Sn) ? sample[((size_t)b * Sn + t0) * 2 + 0]
                               : sample[((size_t)b * Sn + (t0 - Sn)) * 2 + 1];
    const float gy = (t1 < Sn) ? sample[((size_t)b * Sn + t1) * 2 + 0]
                               : sample[((size_t)b * Sn + (t1 - Sn)) * 2 + 1];

    // bilinear, padding_mode='zeros', align_corners=False, H=W=G
    const float ix = ((gx + 1.0f) * (float)Gn - 1.0f) * 0.5f;
    const float iy = ((gy + 1.0f) * (float)Gn - 1.0f) * 0.5f;
    const float x0f = floorf(ix), y0f = floorf(iy);
    const int   x0  = (int)x0f,   y0  = (int)y0f;
    const float wx1 = ix - x0f, wx0 = 1.0f - wx1;
    const float wy1 = iy - y0f, wy0 = 1.0f - wy1;

    const int   xs[2]  = { x0, x0 + 1 };
    const int   ys[2]  = { y0, y0 + 1 };
    const float wxs[2] = { wx0, wx1 };
    const float wys[2] = { wy0, wy1 };

    int   src[4];
    float w[4];
    #pragma unroll
    for (int jy = 0; jy < 2; ++jy)
        #pragma unroll
        for (int jx = 0; jx < 2; ++jx) {
            const int xi = xs[jx], yi = ys[jy];
            const bool valid = (xi >= 0) && (xi < Gn) && (yi >= 0) && (yi < Gn);
            const int xc = min(max(xi, 0), Gn - 1);
            const int yc = min(max(yi, 0), Gn - 1);
            src[jy * 2 + jx] = yc * Gn + xc;          // source sequence index
            w[jy * 2 + jx]   = valid ? wxs[jx] * wys[jy] : 0.0f;
        }

    const float4* q4 = (const float4*)(Q + ((size_t)b * Sn + s) * Dn);
    const float4* k4 = (const float4*)(K + (size_t)b * Sn * Dn);
    const float4* v4 = (const float4*)(V + (size_t)b * Sn * Dn);

    const int c4 = threadIdx.x;                       // 0..191 (Dn/4)
    float skx = 0, sky = 0, skz = 0, skw = 0;
    float svx = 0, svy = 0, svz = 0, svw = 0;
    #pragma unroll
    for (int i = 0; i < 4; ++i) {
        const float4 kv = k4[(size_t)src[i] * (Dn / 4) + c4];
        const float4 vv = v4[(size_t)src[i] * (Dn / 4) + c4];
        skx = fmaf(w[i], kv.x, skx); sky = fmaf(w[i], kv.y, sky);
        skz = fmaf(w[i], kv.z, skz); skw = fmaf(w[i], kv.w, skw);
        svx = fmaf(w[i], vv.x, svx); svy = fmaf(w[i], vv.y, svy);
        svz = fmaf(w[i], vv.z, svz); svw = fmaf(w[i], vv.w, svw);
    }
    const float4 qv = q4[c4];
    float dot = qv.x * skx + qv.y * sky + qv.z * skz + qv.w * skw;

    // block reduction: wave32 shuffle tree, then LDS across 6 waves
    __shared__ float red[6];
    #pragma unroll
    for (int off = 16; off > 0; off >>= 1)
        dot += __shfl_down(dot, off, 32);
    const int lane = threadIdx.x & 31, wv = threadIdx.x >> 5;
    if (lane == 0) red[wv] = dot;
    __syncthreads();
    if (threadIdx.x == 0) {
        float t = 0.0f;
        #pragma unroll
        for (int i = 0; i < 6; ++i) t += red[i];
        red[0] = 1.0f / (1.0f + __expf(-TEMP * t));
    }
    __syncthreads();
    const float sig = red[0];

    float4* o4 = (float4*)(out + ((size_t)b * Sn + s) * Dn);
    float4 ov;
    ov.x = sig * svx; ov.y = sig * svy; ov.z = sig * svz; ov.w = sig * svw;
    o4[c4] = ov;
}

// ---------------------------------------------------------------------------
extern "C" void kernel_launch(void* const* d_in, const int* in_sizes, int n_in,
                              void* d_out, int out_size, void* d_ws, size_t ws_size,
                              hipStream_t stream) {
    // setup_inputs order: x, indices(unused), Wq, bq, Wk, bk, Wv, bv,
    //                     W1, b1, W2, b2, W3, b3
    const float* x  = (const float*)d_in[0];
    const float* Wq = (const float*)d_in[2];
    const float* bq = (const float*)d_in[3];
    const float* Wk = (const float*)d_in[4];
    const float* bk = (const float*)d_in[5];
    const float* Wv = (const float*)d_in[6];
    const float* bv = (const float*)d_in[7];
    const float* W1 = (const float*)d_in[8];
    const float* b1 = (const float*)d_in[9];
    const float* W2 = (const float*)d_in[10];
    const float* b2 = (const float*)d_in[11];
    const float* W3 = (const float*)d_in[12];
    const float* b3 = (const float*)d_in[13];

    // workspace layout: Q | K | V | sample | H1
    const size_t qkv_elems = (size_t)MROWS * Dn;
    float* Q      = (float*)d_ws;
    float* K      = Q + qkv_elems;
    float* V      = K + qkv_elems;
    float* sample = V + qkv_elems;            // MROWS * 2 floats
    float* H1     = sample + (size_t)MROWS * 2; // MROWS * 60 floats

    float* out = (float*)d_out;

    // 1) Q/K/V projections on the fp32 WMMA pipe
    dim3 g1(MROWS / 32, Dn / 128, 3);         // 576 x 6 x 3
    qkv_wmma_kernel<<<g1, 256, 0, stream>>>(x, Wq, bq, Wk, bk, Wv, bv, Q, K, V);

    // 2a) fc1 on the WMMA pipe (bias + relu fused)
    fc1_wmma_kernel<<<MROWS / 32, 256, 0, stream>>>(Q, K, W1, b1, H1);

    // 2b) fc2/relu/fc3/tanh -> sampling coordinates
    mlp_tail_kernel<<<MROWS, 64, 0, stream>>>(H1, W2, b2, W3, b3, sample);

    // 3) bilinear sample + score + scale
    dim3 g3(Sn, Bn, 1);
    sample_attend_kernel<<<g3, 192, 0, stream>>>(Q, K, V, sample, out);
}